// SectGAT_1082331759099
// MI455X (gfx1250) — compile-verified
//
#include <hip/hip_runtime.h>
#include <hip/hip_bf16.h>
#include <math.h>

typedef __attribute__((ext_vector_type(2))) float v2f;
typedef __attribute__((ext_vector_type(8))) float v8f;

#define H 8
#define D 64
#define FEAT 512           // H*D
#define SENT_DIM 256
#define SECT_DIM 512
#define FFN_DIM 2048
#define NEG_SLOPE 0.2f

// ---------------------------------------------------------------------------
// WMMA f32 GEMM:  C[M,N] = act( A[M,K] @ B[K,N] + bias[N] )
// One wave -> one 16x64 output strip (4 accumulators sharing one A-frag per
// K-step: 4 independent WMMA chains, 4x A reuse). 8 waves per block along N
// -> block spans 512 columns. Requires M%16==0, N%512==0, K%4==0 (all true).
// A-frag per ISA 7.12.2: lane<16 -> M=lane, K=k0..k0+1 ; lane>=16 -> K=k0+2..3
// B/C/D row-striped across lanes; D VGPR v -> row v (lanes 0-15) / v+8.
// ---------------------------------------------------------------------------
__global__ void gemm_f32_wmma(const float* __restrict__ A,
                              const float* __restrict__ B,
                              const float* __restrict__ bias,
                              float* __restrict__ C,
                              int M, int N, int K, int act /*0 none,1 relu*/)
{
    const int lane = threadIdx.x & 31;
    const int wave = threadIdx.x >> 5;
    const int hp   = lane >> 4;        // half-wave select
    const int l16  = lane & 15;

    const int col0 = (blockIdx.x * 8 + wave) * 64;   // 16x64 strip per wave
    const int row0 = blockIdx.y * 16;
    if (col0 >= N || row0 >= M) return;

    const float* __restrict__ Arow = A + (size_t)(row0 + l16) * K;

    v8f acc0 = {}, acc1 = {}, acc2 = {}, acc3 = {};
    for (int k0 = 0; k0 < K; k0 += 4) {
        const int ka = k0 + 2 * hp;
        v2f a;
        a.x = Arow[ka];
        a.y = Arow[ka + 1];
        const float* __restrict__ B0 = B + (size_t)ka * N + col0 + l16;
        const float* __restrict__ B1 = B0 + N;
        v2f b0, b1, b2, b3;
        b0.x = B0[0];  b0.y = B1[0];
        b1.x = B0[16]; b1.y = B1[16];
        b2.x = B0[32]; b2.y = B1[32];
        b3.x = B0[48]; b3.y = B1[48];
        acc0 = __builtin_amdgcn_wmma_f32_16x16x4_f32(false, a, false, b0, (short)0, acc0, false, false);
        acc1 = __builtin_amdgcn_wmma_f32_16x16x4_f32(false, a, false, b1, (short)0, acc1, false, false);
        acc2 = __builtin_amdgcn_wmma_f32_16x16x4_f32(false, a, false, b2, (short)0, acc2, false, false);
        acc3 = __builtin_amdgcn_wmma_f32_16x16x4_f32(false, a, false, b3, (short)0, acc3, false, false);
    }

    v8f accs[4] = {acc0, acc1, acc2, acc3};
#pragma unroll
    for (int t = 0; t < 4; ++t) {
        const int c = col0 + t * 16 + l16;
        const float bv = bias ? bias[c] : 0.0f;
#pragma unroll
        for (int v = 0; v < 8; ++v) {
            float x = accs[t][v] + bv;
            if (act == 1) x = fmaxf(x, 0.0f);
            C[(size_t)(row0 + v + 8 * hp) * N + c] = x;
        }
    }
}

// ---------------------------------------------------------------------------
// el[n,h] = sum_d F[n, h*64+d] * a[h,d]   (one wave per (n,h))
// ---------------------------------------------------------------------------
__global__ void row_head_dot(const float* __restrict__ F,
                             const float* __restrict__ a,
                             float* __restrict__ out, int n_rows)
{
    const int gw   = (int)((blockIdx.x * (size_t)blockDim.x + threadIdx.x) >> 5);
    const int lane = threadIdx.x & 31;
    const int n = gw >> 3;
    const int h = gw & 7;
    if (n >= n_rows) return;
    const float* f = F + (size_t)n * FEAT + h * D;
    const float* w = a + h * D;
    float p = f[lane] * w[lane] + f[lane + 32] * w[lane + 32];
#pragma unroll
    for (int off = 16; off; off >>= 1) p += __shfl_down(p, off, 32);
    if (lane == 0) out[n * H + h] = p;
}

__global__ void fill_f32(float* __restrict__ p, long n, float v)
{
    long i = blockIdx.x * (long)blockDim.x + threadIdx.x;
    long stride = (long)gridDim.x * blockDim.x;
    for (; i < n; i += stride) p[i] = v;
}

__device__ inline void atomicMaxFloat(float* addr, float val)
{
    if (val >= 0.0f) atomicMax((int*)addr, __float_as_int(val));
    else             atomicMin((unsigned int*)addr, __float_as_uint(val));
}

// segment max of leaky_relu(el[src]+er[dst]) over dst, per head
__global__ void edge_max(const int* __restrict__ src, const int* __restrict__ dst,
                         const float* __restrict__ el, const float* __restrict__ er,
                         float* __restrict__ m, int E)
{
    const int e = blockIdx.x * blockDim.x + threadIdx.x;
    if (e >= E) return;
    const int s = src[e], d = dst[e];
#pragma unroll
    for (int h = 0; h < H; ++h) {
        float x = el[s * H + h] + er[d * H + h];
        x = (x > 0.0f) ? x : NEG_SLOPE * x;
        atomicMaxFloat(&m[d * H + h], x);
    }
}

__global__ void fix_m(float* __restrict__ m, int n)
{
    const int i = blockIdx.x * blockDim.x + threadIdx.x;
    if (i >= n) return;
    const float x = m[i];
    m[i] = isfinite(x) ? x : 0.0f;
}

// one wave per edge: ex = exp(lrelu(el+er) - m[dst]); denom += ex;
// msg[dst,:] += ex[h(j)] * F[src,:]
__global__ void edge_scatter(const int* __restrict__ src, const int* __restrict__ dst,
                             const float* __restrict__ el, const float* __restrict__ er,
                             const float* __restrict__ m,  const float* __restrict__ F,
                             float* __restrict__ denom, float* __restrict__ msg, int E)
{
    const int gw   = (int)((blockIdx.x * (size_t)blockDim.x + threadIdx.x) >> 5);
    const int lane = threadIdx.x & 31;
    if (gw >= E) return;
    const int s = src[gw], d = dst[gw];

    float exh[H];
#pragma unroll
    for (int h = 0; h < H; ++h) {
        float x = el[s * H + h] + er[d * H + h];
        x = (x > 0.0f) ? x : NEG_SLOPE * x;
        exh[h] = expf(x - m[d * H + h]);
    }
    if (lane < H) atomicAdd(&denom[d * H + lane], exh[lane]);

    const float* f = F + (size_t)s * FEAT;
    float* o = msg + (size_t)d * FEAT;
#pragma unroll
    for (int i = 0; i < FEAT / 32; ++i) {
        const int j = lane + 32 * i;
        atomicAdd(&o[j], exh[j >> 6] * f[j]);
    }
}

// in-place: msg -> elu(msg/max(denom,1e-9) + bias)
__global__ void gat_finalize(float* __restrict__ msg, const float* __restrict__ denom,
                             const float* __restrict__ bias, int n)
{
    const long i = blockIdx.x * (long)blockDim.x + threadIdx.x;
    if (i >= (long)n * FEAT) return;
    const int row = (int)(i >> 9);
    const int j   = (int)(i & (FEAT - 1));
    const float den = fmaxf(denom[row * H + (j >> 6)], 1e-9f);
    const float x = msg[i] / den + bias[j];
    msg[i] = (x > 0.0f) ? x : expm1f(x);
}

// z = sigmoid([s2S,S2S] . fw + fb);  h = sect + z*s2S + (1-z)*S2S
__global__ void fusion_gate(const float* __restrict__ s2S, const float* __restrict__ S2S,
                            const float* __restrict__ sect,
                            const float* __restrict__ fw, const float* __restrict__ fb,
                            float* __restrict__ h, int n)
{
    __shared__ float red[8];
    const int row = blockIdx.x;
    const int t = threadIdx.x;             // 256 threads
    const float* a = s2S + (size_t)row * FEAT;
    const float* b = S2S + (size_t)row * FEAT;

    float p = a[t] * fw[t] + a[t + 256] * fw[t + 256]
            + b[t] * fw[512 + t] + b[t + 256] * fw[768 + t];
#pragma unroll
    for (int off = 16; off; off >>= 1) p += __shfl_down(p, off, 32);
    if ((t & 31) == 0) red[t >> 5] = p;
    __syncthreads();
    if (t == 0) {
        float s = 0.0f;
#pragma unroll
        for (int i = 0; i < 8; ++i) s += red[i];
        red[0] = 1.0f / (1.0f + expf(-(s + fb[0])));
    }
    __syncthreads();
    const float z = red[0];
    float* o = h + (size_t)row * FEAT;
    o[t]       = sect[(size_t)row * FEAT + t]       + z * a[t]       + (1.0f - z) * b[t];
    o[t + 256] = sect[(size_t)row * FEAT + t + 256] + z * a[t + 256] + (1.0f - z) * b[t + 256];
}

// ---------------------------------------------------------------------------
extern "C" void kernel_launch(void* const* d_in, const int* in_sizes, int n_in,
                              void* d_out, int out_size, void* d_ws, size_t ws_size,
                              hipStream_t stream)
{
    const float* sent   = (const float*)d_in[0];
    const float* sect   = (const float*)d_in[1];
    const int*   e1_src = (const int*)d_in[2];
    const int*   e1_dst = (const int*)d_in[3];
    const int*   e2_src = (const int*)d_in[4];
    const int*   e2_dst = (const int*)d_in[5];
    const float* W1_src = (const float*)d_in[6];
    const float* W1_dst = (const float*)d_in[7];
    const float* al1    = (const float*)d_in[8];
    const float* ar1    = (const float*)d_in[9];
    const float* b1     = (const float*)d_in[10];
    const float* W2     = (const float*)d_in[11];
    const float* al2    = (const float*)d_in[12];
    const float* ar2    = (const float*)d_in[13];
    const float* b2     = (const float*)d_in[14];
    const float* fus_w  = (const float*)d_in[15];
    const float* fus_b  = (const float*)d_in[16];
    const float* ffn_w1 = (const float*)d_in[17];
    const float* ffn_b1 = (const float*)d_in[18];
    const float* ffn_w2 = (const float*)d_in[19];
    const float* ffn_b2 = (const float*)d_in[20];
    float* out = (float*)d_out;

    const int N_SENT = in_sizes[0] / SENT_DIM;   // 100000
    const int N_SECT = in_sizes[1] / SECT_DIM;   // 20000
    const int E1 = in_sizes[2];
    const int E2 = in_sizes[4];

    // ---- workspace partition (floats), with aliasing ----
    float* ws = (float*)d_ws;
    size_t off = 0;
    float* fs1  = ws + off; off += (size_t)N_SENT * FEAT;   // aliased later by FFN hidden t
    float* fd1  = ws + off; off += (size_t)N_SECT * FEAT;   // aliased later by fused h
    float* fs2  = ws + off; off += (size_t)N_SECT * FEAT;
    float* msg1 = ws + off; off += (size_t)N_SECT * FEAT;   // becomes s2S in place
    float* msg2 = ws + off; off += (size_t)N_SECT * FEAT;   // becomes S2S in place
    float* el1  = ws + off; off += (size_t)N_SENT * H;
    float* er1  = ws + off; off += (size_t)N_SECT * H;
    float* el2  = ws + off; off += (size_t)N_SECT * H;
    float* er2  = ws + off; off += (size_t)N_SECT * H;
    float* m1   = ws + off; off += (size_t)N_SECT * H;
    float* m2   = ws + off; off += (size_t)N_SECT * H;
    float* den1 = ws + off; off += (size_t)N_SECT * H;
    float* den2 = ws + off; off += (size_t)N_SECT * H;
    float* t_hid = fs1;   // [N_SECT, FFN_DIM] fits inside fs1 region
    float* h_buf = fd1;   // [N_SECT, FEAT]

    const dim3 blk(256);

    // 1) projections (WMMA f32 GEMMs; block spans 512 columns)
    gemm_f32_wmma<<<dim3(FEAT / 512, N_SENT / 16), blk, 0, stream>>>(
        sent, W1_src, nullptr, fs1, N_SENT, FEAT, SENT_DIM, 0);
    gemm_f32_wmma<<<dim3(FEAT / 512, N_SECT / 16), blk, 0, stream>>>(
        sect, W1_dst, nullptr, fd1, N_SECT, FEAT, SECT_DIM, 0);
    gemm_f32_wmma<<<dim3(FEAT / 512, N_SECT / 16), blk, 0, stream>>>(
        sect, W2, nullptr, fs2, N_SECT, FEAT, SECT_DIM, 0);

    // 2) per-head attention coefficients
    {
        int wsent = N_SENT * H, wsect = N_SECT * H;
        row_head_dot<<<(wsent * 32 + 255) / 256, blk, 0, stream>>>(fs1, al1, el1, N_SENT);
        row_head_dot<<<(wsect * 32 + 255) / 256, blk, 0, stream>>>(fd1, ar1, er1, N_SECT);
        row_head_dot<<<(wsect * 32 + 255) / 256, blk, 0, stream>>>(fs2, al2, el2, N_SECT);
        row_head_dot<<<(wsect * 32 + 255) / 256, blk, 0, stream>>>(fs2, ar2, er2, N_SECT);
    }

    // 3) init segment buffers
    const long nh = (long)N_SECT * H;
    const long nf = (long)N_SECT * FEAT;
    fill_f32<<<1024, blk, 0, stream>>>(m1, nh, -INFINITY);
    fill_f32<<<1024, blk, 0, stream>>>(m2, nh, -INFINITY);
    fill_f32<<<1024, blk, 0, stream>>>(den1, nh, 0.0f);
    fill_f32<<<1024, blk, 0, stream>>>(den2, nh, 0.0f);
    fill_f32<<<4096, blk, 0, stream>>>(msg1, nf, 0.0f);
    fill_f32<<<4096, blk, 0, stream>>>(msg2, nf, 0.0f);

    // 4) segment max
    edge_max<<<(E1 + 255) / 256, blk, 0, stream>>>(e1_src, e1_dst, el1, er1, m1, E1);
    edge_max<<<(E2 + 255) / 256, blk, 0, stream>>>(e2_src, e2_dst, el2, er2, m2, E2);
    fix_m<<<(int)((nh + 255) / 256), blk, 0, stream>>>(m1, (int)nh);
    fix_m<<<(int)((nh + 255) / 256), blk, 0, stream>>>(m2, (int)nh);

    // 5) softmax-weighted message scatter (one wave per edge)
    edge_scatter<<<(int)(((size_t)E1 * 32 + 255) / 256), blk, 0, stream>>>(
        e1_src, e1_dst, el1, er1, m1, fs1, den1, msg1, E1);
    edge_scatter<<<(int)(((size_t)E2 * 32 + 255) / 256), blk, 0, stream>>>(
        e2_src, e2_dst, el2, er2, m2, fs2, den2, msg2, E2);

    // 6) normalize + bias + ELU (in place -> s2S / S2S)
    gat_finalize<<<(int)((nf + 255) / 256), blk, 0, stream>>>(msg1, den1, b1, N_SECT);
    gat_finalize<<<(int)((nf + 255) / 256), blk, 0, stream>>>(msg2, den2, b2, N_SECT);

    // 7) fusion gate -> h  (h aliases fd1, which is dead by now)
    fusion_gate<<<N_SECT, blk, 0, stream>>>(msg1, msg2, sect, fus_w, fus_b, h_buf, N_SECT);

    // 8) FFN (WMMA f32 GEMMs, bias+ReLU fused in epilogue)
    gemm_f32_wmma<<<dim3(FFN_DIM / 512, N_SECT / 16), blk, 0, stream>>>(
        h_buf, ffn_w1, ffn_b1, t_hid, N_SECT, FFN_DIM, SECT_DIM, 1);
    gemm_f32_wmma<<<dim3(SECT_DIM / 512, N_SECT / 16), blk, 0, stream>>>(
        t_hid, ffn_w2, ffn_b2, out, N_SECT, SECT_DIM, FFN_DIM, 0);
}